// Loss_Inception_v3_42984032699180
// MI455X (gfx1250) — compile-verified
//
#include <hip/hip_runtime.h>
#include <hip/hip_bf16.h>

// Label-smoothed cross-entropy, B=4096 rows x K=32000 fp32 logits -> scalar.
// Memory-bound: 512 MB single pass => ~22us floor at 23.3 TB/s.
// CDNA5 async global->LDS double-buffered stream, 2x-unrolled software pipeline
// with dual accumulator sets for ILP through the v_exp_f32 chain.

#define K_DIM      32000
#define NTHREADS   320                      // 10 wave32s
#define CHUNK_FL   (NTHREADS * 4)           // 1280 floats per chunk
#define CHUNK_BY   (CHUNK_FL * 4)           // 5120 bytes
#define NCHUNK     (K_DIM / CHUNK_FL)       // exactly 25
#define NWAVES     (NTHREADS / 32)          // 10
#define SMOOTH     0.1f
#define L2E        1.44269504088896340736f  // log2(e)
#define LN2        0.69314718055994530942f  // ln(2)

// Issue one 16B/lane async copy into LDS. s_wait_dscnt 0 first closes the
// (theoretical) WAR hazard: DS reads and async LDS writes use different
// counters and are unordered, so make sure our prior ds_load of this buffer
// has completed before the TDM-side write can land.
#define ASYNC_ISSUE(ldsoff, gptr)                                   \
    asm volatile("s_wait_dscnt 0x0\n\t"                             \
                 "global_load_async_to_lds_b128 %0, %1, off"        \
                 :: "v"(ldsoff), "v"(gptr) : "memory")

#define ASYNC_WAIT1() asm volatile("s_wait_asynccnt 0x1" ::: "memory")
#define ASYNC_WAIT0() asm volatile("s_wait_asynccnt 0x0" ::: "memory")

// Online logsumexp update for one float4 (5 hw exp2 per 4 elements).
__device__ __forceinline__ void accum4(const float4 x, float& m, float& s, float& ps) {
    ps += (x.x + x.y) + (x.z + x.w);
    const float m4 = fmaxf(fmaxf(x.x, x.y), fmaxf(x.z, x.w));
    const float mn = fmaxf(m, m4);
    s = s * __builtin_amdgcn_exp2f((m   - mn) * L2E)
      +     __builtin_amdgcn_exp2f((x.x - mn) * L2E)
      +     __builtin_amdgcn_exp2f((x.y - mn) * L2E)
      +     __builtin_amdgcn_exp2f((x.z - mn) * L2E)
      +     __builtin_amdgcn_exp2f((x.w - mn) * L2E);
    m = mn;
}

// Merge two (max, scaled-sum) partials.
__device__ __forceinline__ void merge(float& m, float& s, const float mo, const float so) {
    const float mn = fmaxf(m, mo);
    s = s  * __builtin_amdgcn_exp2f((m  - mn) * L2E)
      + so * __builtin_amdgcn_exp2f((mo - mn) * L2E);
    m = mn;
}

__global__ __launch_bounds__(NTHREADS)
void row_loss_kernel(const int* __restrict__ y,
                     const float* __restrict__ p,
                     float* __restrict__ row_out) {
    __shared__ float lbuf[2 * CHUNK_FL];     // 10 KB double buffer
    __shared__ float red_m[NWAVES], red_s[NWAVES], red_p[NWAVES];

    const int tid = threadIdx.x;
    const int row = blockIdx.x;
    const float* rowp = p + (size_t)row * K_DIM;

    // Loop-invariant LDS byte offsets (low 32 bits of flat shared addr are the
    // workgroup-relative LDS offset; hardware adds LDS_BASE).
    const unsigned l0 = (unsigned)(size_t)(void*)lbuf + (unsigned)(tid * 16);
    const unsigned l1 = l0 + CHUNK_BY;

    // Per-lane LDS read pointers (each lane reads back only its own 16 B, so
    // no workgroup barrier is needed; per-wave ASYNCcnt ordering suffices).
    const float4* lp0 = reinterpret_cast<const float4*>(lbuf) + tid;
    const float4* lp1 = lp0 + NTHREADS;

    // ---- prologue: chunk 0 -> buf0 ----
    const float* g = rowp + tid * 4;
    ASYNC_ISSUE(l0, g);
    g += CHUNK_FL;

    float m0 = -3.0e38f, s0 = 0.0f, ps0 = 0.0f;   // even chunks
    float m1 = -3.0e38f, s1 = 0.0f, ps1 = 0.0f;   // odd chunks

    // 12 double-iterations cover chunks 0..23; chunk 24 in epilogue.
    #pragma unroll 1
    for (int it = 0; it < (NCHUNK - 1) / 2; ++it) {
        ASYNC_ISSUE(l1, g); g += CHUNK_FL;        // chunk 2it+1 -> buf1
        ASYNC_WAIT1();                            // chunk 2it ready in buf0
        accum4(*lp0, m0, s0, ps0);

        ASYNC_ISSUE(l0, g); g += CHUNK_FL;        // chunk 2it+2 -> buf0
        ASYNC_WAIT1();                            // chunk 2it+1 ready in buf1
        accum4(*lp1, m1, s1, ps1);
    }
    // ---- epilogue: chunk 24 in buf0 ----
    ASYNC_WAIT0();
    accum4(*lp0, m0, s0, ps0);

    // Merge even/odd accumulator sets.
    merge(m0, s0, m1, s1);
    float psum = ps0 + ps1;

    // ---- wave32 reduction ----
    #pragma unroll
    for (int off = 16; off > 0; off >>= 1) {
        const float mo = __shfl_xor(m0,   off, 32);
        const float so = __shfl_xor(s0,   off, 32);
        const float po = __shfl_xor(psum, off, 32);
        merge(m0, s0, mo, so);
        psum += po;
    }

    const int lane = tid & 31, wid = tid >> 5;
    if (lane == 0) { red_m[wid] = m0; red_s[wid] = s0; red_p[wid] = psum; }
    __syncthreads();

    if (tid == 0) {
        float M = red_m[0], S = red_s[0], P = red_p[0];
        #pragma unroll
        for (int w = 1; w < NWAVES; ++w) {
            merge(M, S, red_m[w], red_s[w]);
            P += red_p[w];
        }
        const float lse = M + LN2 * __builtin_amdgcn_logf(S);   // natural-log LSE
        const float tgt = rowp[y[row]];                          // target logit
        const float loss = (SMOOTH / (float)K_DIM) * (P - (float)K_DIM * lse)
                         + (1.0f - SMOOTH) * (tgt - lse);
        row_out[row] = loss;
    }
}

__global__ __launch_bounds__(256)
void final_reduce_kernel(const float* __restrict__ rl,
                         float* __restrict__ out, int n) {
    __shared__ float sm[8];
    float acc = 0.0f;
    for (int i = threadIdx.x; i < n; i += 256) acc += rl[i];
    #pragma unroll
    for (int off = 16; off > 0; off >>= 1) acc += __shfl_xor(acc, off, 32);
    const int lane = threadIdx.x & 31, wid = threadIdx.x >> 5;
    if (lane == 0) sm[wid] = acc;
    __syncthreads();
    if (threadIdx.x == 0) {
        float t = 0.0f;
        #pragma unroll
        for (int w = 0; w < 8; ++w) t += sm[w];
        out[0] = t;
    }
}

extern "C" void kernel_launch(void* const* d_in, const int* in_sizes, int n_in,
                              void* d_out, int out_size, void* d_ws, size_t ws_size,
                              hipStream_t stream) {
    (void)n_in; (void)out_size; (void)ws_size;
    const int*   y = (const int*)d_in[0];     // targets, B elements
    const float* p = (const float*)d_in[1];   // logits, B*K elements
    const int    B = in_sizes[0];             // 4096 rows

    float* row_losses = (float*)d_ws;         // B floats of scratch

    row_loss_kernel<<<B, NTHREADS, 0, stream>>>(y, p, row_losses);
    final_reduce_kernel<<<1, 256, 0, stream>>>(row_losses, (float*)d_out, B);
}